// WeightLayer_62852551410336
// MI455X (gfx1250) — compile-verified
//
#include <hip/hip_runtime.h>
#include <math.h>

typedef float v2f __attribute__((ext_vector_type(2)));
typedef float v8f __attribute__((ext_vector_type(8)));

#define WMMA_F32_4(A, B, C) \
  __builtin_amdgcn_wmma_f32_16x16x4_f32(false, (A), false, (B), (short)0, (C), false, false)

constexpr int BATCH = 4;
constexpr int NQ    = 8192;   // queries per batch
constexpr int MP    = 8192;   // original points per batch
constexpr int FEAT  = 64;
constexpr int KNN   = 16;
constexpr int K4    = 4;
constexpr float EPS = 1e-5f;

// ---------------------------------------------------------------------------
// Kernel 1: KNN via WMMA distance tiles.
// One wave (32 lanes) owns 16 queries. Score s = |p|^2 - 2 p.q (same ranking
// as full d2). Folded into a single V_WMMA_F32_16X16X4_F32 per 16-point tile:
//   A[m][k] = {px, py, pz, |p|^2}   (16 points  x 4)
//   B[k][n] = {-2qx,-2qy,-2qz, 1 }  (4 x 16 queries)
// D layout: lane j holds query col j (j = lane&15), rows v (+8 if lane>=16).
// Lanes j and j+16 each keep a register top-16 over their 4096 rows, merged
// in LDS with an ascending stable selection (matches jax.lax.top_k order).
// ---------------------------------------------------------------------------
__global__ __launch_bounds__(32)
void knn_kernel(const float* __restrict__ opts,   // (B,3,M)
                const float* __restrict__ qpts,   // (B,3,N)
                int* __restrict__ knn_out)        // (B,N,K)
{
  const int tile = blockIdx.x;                 // B * (N/16) tiles
  const int b    = tile / (NQ / 16);
  const int n0   = (tile % (NQ / 16)) * 16;
  const int lane = threadIdx.x;
  const int lj   = lane & 15;
  const bool hi  = lane >= 16;

  const float* qb = qpts + (size_t)b * 3 * NQ;
  const float* ob = opts + (size_t)b * 3 * MP;

  // B operand (queries), loaded once.
  v2f bop;
  {
    const int n = n0 + lj;
    if (!hi) {
      bop.x = -2.0f * qb[0 * NQ + n];   // K0
      bop.y = -2.0f * qb[1 * NQ + n];   // K1
    } else {
      bop.x = -2.0f * qb[2 * NQ + n];   // K2
      bop.y = 1.0f;                     // K3
    }
  }

  // Register-resident top-16 (unsorted, replace-current-max).
  float bd[16];
  int   bi_[16];
#pragma unroll
  for (int i = 0; i < 16; ++i) { bd[i] = __builtin_inff(); bi_[i] = 0x7fffffff; }
  float worst = __builtin_inff();
  int   wslot = 0;

  for (int m0 = 0; m0 < MP; m0 += 16) {
    const int m = m0 + lj;
    // Prefetch next point tile (L2 stream hint -> global_prefetch_b8).
    if (m0 + 16 < MP) __builtin_prefetch(&ob[0 * MP + m + 16], 0, 1);

    v2f aop;
    if (!hi) {
      aop.x = ob[0 * MP + m];                 // K0 = px
      aop.y = ob[1 * MP + m];                 // K1 = py
    } else {
      const float x = ob[0 * MP + m];
      const float y = ob[1 * MP + m];
      const float z = ob[2 * MP + m];
      aop.x = z;                              // K2 = pz
      aop.y = x * x + y * y + z * z;          // K3 = |p|^2
    }

    v8f d = {0.f, 0.f, 0.f, 0.f, 0.f, 0.f, 0.f, 0.f};
    d = WMMA_F32_4(aop, bop, d);

    const int rbase = m0 + (hi ? 8 : 0);
#pragma unroll
    for (int v = 0; v < 8; ++v) {
      const float s = d[v];
      if (s < worst) {
        const int mi = rbase + v;
#pragma unroll
        for (int i = 0; i < 16; ++i)
          if (i == wslot) { bd[i] = s; bi_[i] = mi; }
        worst = bd[0]; wslot = 0;
#pragma unroll
        for (int i = 1; i < 16; ++i)
          if (bd[i] > worst) { worst = bd[i]; wslot = i; }
      }
    }
  }

  // Merge lanes j / j+16 per query and emit ascending-sorted indices.
  __shared__ float md[16][32];
  __shared__ int   mi[16][32];
  {
    const int base = hi ? 16 : 0;
#pragma unroll
    for (int i = 0; i < 16; ++i) { md[lj][base + i] = bd[i]; mi[lj][base + i] = bi_[i]; }
  }
  __syncthreads();

  if (!hi) {
    const int q = lj;
    int* outp = knn_out + ((size_t)b * NQ + (n0 + q)) * KNN;
    for (int r = 0; r < KNN; ++r) {
      float best = __builtin_inff();
      int   bidx = 0x7fffffff;
      int   bslot = 0;
      for (int e = 0; e < 32; ++e) {
        const float dv = md[q][e];
        const int   iv = mi[q][e];
        if (dv < best || (dv == best && iv < bidx)) { best = dv; bidx = iv; bslot = e; }
      }
      outp[r] = bidx;
      md[q][bslot] = __builtin_inff();
      mi[q][bslot] = 0x7fffffff;
    }
  }
}

// ---------------------------------------------------------------------------
// Kernel 2: fused MLP pipeline. One wave handles 4 queries.
// Layer0 (inner=3, pad K=4) and Layer1 (inner=64 -> 16 chunks of 4) both use
// V_WMMA_F32_16X16X4_F32 with BN folded into per-row affine (y = a*x + c).
// ---------------------------------------------------------------------------
__global__ __launch_bounds__(32)
void mlp_kernel(const float* __restrict__ opts,  const float* __restrict__ qpts,
                const float* __restrict__ lfeat,
                const float* __restrict__ w0, const float* __restrict__ b0,
                const float* __restrict__ g0, const float* __restrict__ be0,
                const float* __restrict__ m0_, const float* __restrict__ v0,
                const float* __restrict__ w1, const float* __restrict__ b1,
                const float* __restrict__ g1, const float* __restrict__ be1,
                const float* __restrict__ m1_, const float* __restrict__ v1,
                const float* __restrict__ w2, const float* __restrict__ b2,
                const int* __restrict__ knn, float* __restrict__ out)
{
  const int grp  = blockIdx.x;                 // B * (N/4) groups
  const int b    = grp / (NQ / 4);
  const int n0   = (grp % (NQ / 4)) * 4;
  const int lane = threadIdx.x;
  const int lj   = lane & 15;
  const bool hi  = lane >= 16;

  __shared__ float sF0[64][65];   // feat0: 64 feats x (4 queries * 16 k)
  __shared__ float sRF[64][17];   // relative_feat: 64 feats x (4 q * 4 k')
  __shared__ float sFG[64][4];    // feat_g
  __shared__ float sA0[64], sC0[64], sA1[64], sC1[64];
  __shared__ float sW2[128];
  __shared__ int   sIdx[4][16];
  __shared__ float sPart[32];
  __shared__ float sW[16];

  // Fold BN into affine; stage small params in LDS.
  for (int r = lane; r < 64; r += 32) {
    const float a0 = g0[r] * rsqrtf(v0[r] + EPS);
    sA0[r] = a0;
    sC0[r] = (b0[r] - m0_[r]) * a0 + be0[r];
    const float a1 = g1[r] * rsqrtf(v1[r] + EPS);
    sA1[r] = a1;
    sC1[r] = (b1[r] - m1_[r]) * a1 + be1[r];
  }
  for (int r = lane; r < 128; r += 32) sW2[r] = w2[r];
  {
    const int* kb = knn + ((size_t)b * NQ + n0) * KNN;
    for (int t = lane; t < 64; t += 32) sIdx[t >> 4][t & 15] = kb[t];
  }
  __syncthreads();

  const float* ob = opts + (size_t)b * 3 * MP;
  const float* qb = qpts + (size_t)b * 3 * NQ;

  // --- layer 0: feat0 = relu(aff0(W0 x rel))  (64 x 64 output) ---
  for (int tc = 0; tc < 4; ++tc) {
    const int q    = tc;
    const int n    = n0 + q;
    const int mIdx = sIdx[q][lj];
    v2f bop;
    if (!hi) {
      bop.x = ob[0 * MP + mIdx] - qb[0 * NQ + n];   // K0 = rel.x
      bop.y = ob[1 * MP + mIdx] - qb[1 * NQ + n];   // K1 = rel.y
    } else {
      bop.x = ob[2 * MP + mIdx] - qb[2 * NQ + n];   // K2 = rel.z
      bop.y = 0.0f;                                 // K3 pad
    }
    for (int tr = 0; tr < 4; ++tr) {
      const int row = tr * 16 + lj;
      v2f aop;
      if (!hi) { aop.x = w0[row * 3 + 0]; aop.y = w0[row * 3 + 1]; }
      else     { aop.x = w0[row * 3 + 2]; aop.y = 0.0f; }
      v8f d = {0.f, 0.f, 0.f, 0.f, 0.f, 0.f, 0.f, 0.f};
      d = WMMA_F32_4(aop, bop, d);
      const int rb = tr * 16 + (hi ? 8 : 0);
#pragma unroll
      for (int v = 0; v < 8; ++v) {
        const int rr = rb + v;
        sF0[rr][tc * 16 + lj] = fmaxf(sA0[rr] * d[v] + sC0[rr], 0.0f);
      }
    }
  }
  __syncthreads();

  // --- feat_g = max over K per (feat, query) ---
#pragma unroll
  for (int i = 0; i < 8; ++i) {
    const int t  = i * 32 + lane;       // 0..255
    const int rr = t >> 2, q = t & 3;
    float mx = sF0[rr][q * 16 + 0];
    for (int k = 1; k < 16; ++k) mx = fmaxf(mx, sF0[rr][q * 16 + k]);
    sFG[rr][q] = mx;
  }
  __syncthreads();

  // --- layer 1: relative_feat = relu(aff1(W1 x feat0[:, first 4 k])) ---
  const int c  = lj;                        // output col: q = c>>2, k' = c&3
  const int sc = (c >> 2) * 16 + (c & 3);   // source col in feat0
  for (int tr = 0; tr < 4; ++tr) {
    v8f acc = {0.f, 0.f, 0.f, 0.f, 0.f, 0.f, 0.f, 0.f};
    for (int t = 0; t < 16; ++t) {
      const int row = tr * 16 + lj;
      const int kb2 = 4 * t + (hi ? 2 : 0);
      v2f aop; aop.x = w1[row * 64 + kb2]; aop.y = w1[row * 64 + kb2 + 1];
      v2f bop; bop.x = sF0[kb2][sc];       bop.y = sF0[kb2 + 1][sc];
      acc = WMMA_F32_4(aop, bop, acc);
    }
    const int rb = tr * 16 + (hi ? 8 : 0);
#pragma unroll
    for (int v = 0; v < 8; ++v) {
      const int rr = rb + v;
      sRF[rr][c] = fmaxf(sA1[rr] * acc[v] + sC1[rr], 0.0f);
    }
  }
  __syncthreads();

  // --- gate: weight = sigmoid(w2 . [relative_feat; feat_g] + b2) ---
  {
    const int h = hi ? 1 : 0;
    const int q = lj >> 2;
    float s = 0.0f;
    for (int f = h * 32; f < h * 32 + 32; ++f)
      s += sW2[f] * sRF[f][lj] + sW2[64 + f] * sFG[f][q];
    sPart[lane] = s;
  }
  __syncthreads();
  if (!hi) {
    const float x = sPart[lj] + sPart[lj + 16] + b2[0];
    sW[lj] = 1.0f / (1.0f + expf(-x));
  }
  __syncthreads();

  // --- output: sum over k' of (1-w)*relative_feat + w*patch_feat ---
  const float* lb = lfeat + (size_t)b * FEAT * MP;
#pragma unroll
  for (int i = 0; i < 8; ++i) {
    const int t = i * 32 + lane;        // 0..255
    const int f = t >> 2, q = t & 3;
    float s = 0.0f;
    for (int k = 0; k < K4; ++k) {
      const int cc = q * 4 + k;
      const float w  = sW[cc];
      const float rf = sRF[f][cc];
      const float pf = lb[(size_t)f * MP + sIdx[q][k]];
      s += (1.0f - w) * rf + w * pf;
    }
    out[((size_t)b * FEAT + f) * NQ + (n0 + q)] = s;
  }
}

extern "C" void kernel_launch(void* const* d_in, const int* in_sizes, int n_in,
                              void* d_out, int out_size, void* d_ws, size_t ws_size,
                              hipStream_t stream) {
  const float* opts  = (const float*)d_in[0];
  const float* qpts  = (const float*)d_in[1];
  const float* lfeat = (const float*)d_in[2];
  const float* w0  = (const float*)d_in[3];
  const float* b0  = (const float*)d_in[4];
  const float* g0  = (const float*)d_in[5];
  const float* be0 = (const float*)d_in[6];
  const float* m0  = (const float*)d_in[7];
  const float* v0  = (const float*)d_in[8];
  const float* w1  = (const float*)d_in[9];
  const float* b1  = (const float*)d_in[10];
  const float* g1  = (const float*)d_in[11];
  const float* be1 = (const float*)d_in[12];
  const float* m1  = (const float*)d_in[13];
  const float* v1  = (const float*)d_in[14];
  const float* w2  = (const float*)d_in[15];
  const float* b2  = (const float*)d_in[16];
  float* out = (float*)d_out;

  int* knn = (int*)d_ws;   // B*N*K ints = 2 MB scratch

  knn_kernel<<<BATCH * (NQ / 16), 32, 0, stream>>>(opts, qpts, knn);
  mlp_kernel<<<BATCH * (NQ / 4), 32, 0, stream>>>(opts, qpts, lfeat,
                                                  w0, b0, g0, be0, m0, v0,
                                                  w1, b1, g1, be1, m1, v1,
                                                  w2, b2, knn, out);
}